// GSageInc_6073083756548
// MI455X (gfx1250) — compile-verified
//
#include <hip/hip_runtime.h>
#include <hip/hip_bf16.h>

typedef __attribute__((ext_vector_type(2))) float v2f;
typedef __attribute__((ext_vector_type(8))) float v8f;

#define D 64
#define NCLS 8

// ---------------------------------------------------------------- zero
__global__ void zero_f32(float* __restrict__ p, long n) {
  long i = (long)blockIdx.x * blockDim.x + threadIdx.x;
  if (i < n) p[i] = 0.0f;
}

// ---------------------------------------------------------------- degree
__global__ void deg_count(const int* __restrict__ tgt, float* __restrict__ deg, int E_) {
  int i = blockIdx.x * blockDim.x + threadIdx.x;
  if (i < E_) atomicAdd(&deg[tgt[i]], 1.0f);
}

__global__ void inv_deg(float* __restrict__ deg, int n) {
  int i = blockIdx.x * blockDim.x + threadIdx.x;
  if (i < n) {
    float d = deg[i];
    deg[i] = (d > 0.0f) ? (1.0f / d) : 0.0f;   // deg>0 implies deg>=1
  }
}

// ---------------------------------------------------------------- scatter (sum over edges)
__global__ void scatter_add(const float* __restrict__ X, const int* __restrict__ src,
                            const int* __restrict__ tgt, float* __restrict__ agg,
                            const int* __restrict__ lidx, int layer_i, int always,
                            long total) {
  if (!(always || layer_i <= *lidx)) return;   // uniform scalar branch
  long tid = (long)blockIdx.x * blockDim.x + threadIdx.x;
  if (tid >= total) return;
  int e = (int)(tid >> 6);
  int f = (int)(tid & 63);
  int s = src[e];
  int t = tgt[e];
  atomicAdd(&agg[(size_t)t * D + f], X[(size_t)s * D + f]);
}

// ---------------------------------------------------------------- fused SAGE GEMM + bias + ReLU + BN stats
// out[16x64 tile] = (agg*inv_deg) @ Wl + X @ Wr + b ; relu ; accumulate sum/sumsq per feature
__global__ __launch_bounds__(256) void sage_gemm(
    const float* __restrict__ X, const float* __restrict__ agg,
    const float* __restrict__ invd, const float* __restrict__ Wl,
    const float* __restrict__ Wr, const float* __restrict__ bias,
    float* __restrict__ outp, float* __restrict__ stats,
    const int* __restrict__ lidx, int layer_i, int always, int n_nodes) {
  if (!(always || layer_i <= *lidx)) return;   // whole wave exits together (EXEC rule ok)
  const int wave = threadIdx.x >> 5;
  const int t    = threadIdx.x & 31;
  const int half = t >> 4;
  const int tm   = t & 15;
  const int row0 = (blockIdx.x * 8 + wave) * 16;
  if (row0 >= n_nodes) return;

  const float idg = invd[row0 + tm];
  const float* __restrict__ xrow = X   + (size_t)(row0 + tm) * D;
  const float* __restrict__ arow = agg + (size_t)(row0 + tm) * D;

  v8f acc[4];
  acc[0] = (v8f){}; acc[1] = (v8f){}; acc[2] = (v8f){}; acc[3] = (v8f){};

#pragma unroll
  for (int k0 = 0; k0 < D; k0 += 4) {
    const int kk = k0 + 2 * half;
    v2f aL, aR;
    aL.x = arow[kk] * idg;  aL.y = arow[kk + 1] * idg;   // A1 = agg * inv_deg
    aR.x = xrow[kk];        aR.y = xrow[kk + 1];         // A2 = root features
    const float* __restrict__ wl = Wl + (size_t)kk * D;
    const float* __restrict__ wr = Wr + (size_t)kk * D;
#pragma unroll
    for (int nt = 0; nt < 4; ++nt) {
      const int n = nt * 16 + tm;
      v2f bl; bl.x = wl[n]; bl.y = wl[D + n];
      v2f br; br.x = wr[n]; br.y = wr[D + n];
      acc[nt] = __builtin_amdgcn_wmma_f32_16x16x4_f32(false, aL, false, bl,
                                                      (short)0, acc[nt], false, false);
      acc[nt] = __builtin_amdgcn_wmma_f32_16x16x4_f32(false, aR, false, br,
                                                      (short)0, acc[nt], false, false);
    }
  }

  // epilogue: bias + relu + store + BN partial stats (each lane owns one feature column)
#pragma unroll
  for (int nt = 0; nt < 4; ++nt) {
    const int n  = nt * 16 + tm;
    const float bv = bias[n];
    float s = 0.0f, sq = 0.0f;
#pragma unroll
    for (int r = 0; r < 8; ++r) {
      float v = acc[nt][r] + bv;
      v = v > 0.0f ? v : 0.0f;
      const int row = row0 + half * 8 + r;
      outp[(size_t)row * D + n] = v;
      s  += v;
      sq += v * v;
    }
    atomicAdd(&stats[n],      s);
    atomicAdd(&stats[64 + n], sq);
  }
}

// ---------------------------------------------------------------- BN apply (or passthrough when layer skipped)
__global__ void bn_apply(const float* __restrict__ tin, const float* __restrict__ xpass,
                         const float* __restrict__ g, const float* __restrict__ beta,
                         float* __restrict__ outp, const float* __restrict__ stats,
                         const int* __restrict__ lidx, int layer_i, int always,
                         float inv_n, long total) {
  long i = (long)blockIdx.x * blockDim.x + threadIdx.x;
  if (i >= total) return;
  if (!(always || layer_i <= *lidx)) { outp[i] = xpass[i]; return; }
  int n = (int)(i & 63);
  float mu  = stats[n] * inv_n;
  float var = stats[64 + n] * inv_n - mu * mu;
  var = var > 0.0f ? var : 0.0f;
  float rs = rsqrtf(var + 1e-5f);
  outp[i] = g[n] * (tin[i] - mu) * rs + beta[n];
}

// ---------------------------------------------------------------- final SAGE 64->8 (zero-padded N tile)
__global__ __launch_bounds__(256) void final_gemm(
    const float* __restrict__ X, const float* __restrict__ agg,
    const float* __restrict__ invd, const float* __restrict__ Wl,
    const float* __restrict__ Wr, const float* __restrict__ bias,
    float* __restrict__ logits, int n_nodes) {
  const int wave = threadIdx.x >> 5;
  const int t    = threadIdx.x & 31;
  const int half = t >> 4;
  const int tm   = t & 15;
  const int row0 = (blockIdx.x * 8 + wave) * 16;
  if (row0 >= n_nodes) return;

  const float idg = invd[row0 + tm];
  const float* __restrict__ xrow = X   + (size_t)(row0 + tm) * D;
  const float* __restrict__ arow = agg + (size_t)(row0 + tm) * D;
  const bool ok = (tm < NCLS);
  const int nidx = ok ? tm : 0;    // clamp so every lane issues a valid load

  v8f acc = (v8f){};
#pragma unroll
  for (int k0 = 0; k0 < D; k0 += 4) {
    const int kk = k0 + 2 * half;
    v2f aL, aR;
    aL.x = arow[kk] * idg;  aL.y = arow[kk + 1] * idg;
    aR.x = xrow[kk];        aR.y = xrow[kk + 1];
    v2f bl, br;
    float bl0 = Wl[(size_t)kk * NCLS + nidx];
    float bl1 = Wl[(size_t)(kk + 1) * NCLS + nidx];
    float br0 = Wr[(size_t)kk * NCLS + nidx];
    float br1 = Wr[(size_t)(kk + 1) * NCLS + nidx];
    bl.x = ok ? bl0 : 0.0f;  bl.y = ok ? bl1 : 0.0f;
    br.x = ok ? br0 : 0.0f;  br.y = ok ? br1 : 0.0f;
    acc = __builtin_amdgcn_wmma_f32_16x16x4_f32(false, aL, false, bl,
                                                (short)0, acc, false, false);
    acc = __builtin_amdgcn_wmma_f32_16x16x4_f32(false, aR, false, br,
                                                (short)0, acc, false, false);
  }

  if (ok) {                       // divergence only after all WMMA ops
    const float bv = bias[tm];
#pragma unroll
    for (int r = 0; r < 8; ++r) {
      const int row = row0 + half * 8 + r;
      logits[(size_t)row * NCLS + tm] = acc[r] + bv;
    }
  }
}

// ---------------------------------------------------------------- row-wise log_softmax over 8 classes
__global__ void logsoftmax8(const float* __restrict__ in, float* __restrict__ out, int n) {
  int r = blockIdx.x * blockDim.x + threadIdx.x;
  if (r >= n) return;
  float v[NCLS];
  float m = -3.402823e38f;
#pragma unroll
  for (int j = 0; j < NCLS; ++j) { v[j] = in[(size_t)r * NCLS + j]; m = fmaxf(m, v[j]); }
  float s = 0.0f;
#pragma unroll
  for (int j = 0; j < NCLS; ++j) s += expf(v[j] - m);
  float l = logf(s);
#pragma unroll
  for (int j = 0; j < NCLS; ++j) out[(size_t)r * NCLS + j] = v[j] - m - l;
}

// ================================================================ launch
extern "C" void kernel_launch(void* const* d_in, const int* in_sizes, int n_in,
                              void* d_out, int out_size, void* d_ws, size_t ws_size,
                              hipStream_t stream) {
  const float* x        = (const float*)d_in[0];
  const int*   ei       = (const int*)d_in[1];
  const int*   lidx     = (const int*)d_in[2];
  const float* proj_Wl  = (const float*)d_in[3];
  const float* proj_Wr  = (const float*)d_in[4];
  const float* proj_b   = (const float*)d_in[5];
  const float* norm_g   = (const float*)d_in[6];
  const float* norm_b   = (const float*)d_in[7];
  const float* conv_Wl  = (const float*)d_in[8];
  const float* conv_Wr  = (const float*)d_in[9];
  const float* conv_b   = (const float*)d_in[10];
  const float* conv_g   = (const float*)d_in[11];
  const float* conv_be  = (const float*)d_in[12];
  const float* fin_Wl   = (const float*)d_in[13];
  const float* fin_Wr   = (const float*)d_in[14];
  const float* fin_b    = (const float*)d_in[15];

  const int Nn = in_sizes[0] / D;
  const int Ee = in_sizes[1] / 2;
  const int* src = ei;
  const int* tgt = ei + Ee;

  float* ws     = (float*)d_ws;
  float* deg    = ws;                              // Nn (becomes inv_deg in place)
  float* agg    = deg   + Nn;                      // Nn*64
  float* stats  = agg   + (size_t)Nn * D;          // 128 (sum | sumsq)
  float* tmp    = stats + 128;                     // Nn*64 (pre-BN)
  float* h      = tmp   + (size_t)Nn * D;          // Nn*64 (hidden state, in-place)
  float* logits = h     + (size_t)Nn * D;          // Nn*8

  const int TB = 256;
  const long hN      = (long)Nn * D;
  const long aggN    = hN + 128;
  const long scatT   = (long)Ee * D;
  const int  gScat   = (int)((scatT + TB - 1) / TB);
  const int  gElem   = (int)((hN + TB - 1) / TB);
  const int  gAgg    = (int)((aggN + TB - 1) / TB);
  const int  gNode   = (Nn + TB - 1) / TB;
  const int  gGemm   = (Nn + 127) / 128;           // 8 waves/block * 16 rows/wave
  const float inv_n  = 1.0f / (float)Nn;

  // degrees
  zero_f32<<<gNode, TB, 0, stream>>>(deg, Nn);
  deg_count<<<(Ee + TB - 1) / TB, TB, 0, stream>>>(tgt, deg, Ee);
  inv_deg<<<gNode, TB, 0, stream>>>(deg, Nn);

  // proj layer (always active): h = BN(relu(sage(x)))
  zero_f32<<<gAgg, TB, 0, stream>>>(agg, aggN);
  scatter_add<<<gScat, TB, 0, stream>>>(x, src, tgt, agg, lidx, 0, 1, scatT);
  sage_gemm<<<gGemm, TB, 0, stream>>>(x, agg, deg, proj_Wl, proj_Wr, proj_b,
                                      tmp, stats, lidx, 0, 1, Nn);
  bn_apply<<<gElem, TB, 0, stream>>>(tmp, x, norm_g, norm_b, h, stats,
                                     lidx, 0, 1, inv_n, hN);

  // 3 conv layers, gated on device-side layer_idx
  for (int i = 0; i < 3; ++i) {
    zero_f32<<<gAgg, TB, 0, stream>>>(agg, aggN);
    scatter_add<<<gScat, TB, 0, stream>>>(h, src, tgt, agg, lidx, i, 0, scatT);
    sage_gemm<<<gGemm, TB, 0, stream>>>(h, agg, deg,
                                        conv_Wl + (size_t)i * D * D,
                                        conv_Wr + (size_t)i * D * D,
                                        conv_b  + (size_t)i * D,
                                        tmp, stats, lidx, i, 0, Nn);
    bn_apply<<<gElem, TB, 0, stream>>>(tmp, h,
                                       conv_g  + (size_t)i * D,
                                       conv_be + (size_t)i * D,
                                       h, stats, lidx, i, 0, inv_n, hN);
  }

  // final layer + log_softmax
  zero_f32<<<gAgg, TB, 0, stream>>>(agg, aggN);
  scatter_add<<<gScat, TB, 0, stream>>>(h, src, tgt, agg, lidx, 0, 1, scatT);
  final_gemm<<<gGemm, TB, 0, stream>>>(h, agg, deg, fin_Wl, fin_Wr, fin_b, logits, Nn);
  logsoftmax8<<<gNode, TB, 0, stream>>>(logits, (float*)d_out, Nn);
}